// Router_24240795419343
// MI455X (gfx1250) — compile-verified
//
#include <hip/hip_runtime.h>
#include <hip/hip_bf16.h>
#include <cfloat>

// ---------------------------------------------------------------------------
// MoE router for MI455X (gfx1250, wave32).
//   logits = x @ W^T           (M=32768, N=64, K=1024, f32)
//   top-2 + softmax gates, full-softmax mean -> load-balance loss
// Bandwidth-bound: 134 MB of x @ 23.3 TB/s ~= 5.8 us floor; f32 WMMA is free.
// ---------------------------------------------------------------------------

typedef __attribute__((ext_vector_type(2))) float v2f;
typedef __attribute__((ext_vector_type(8))) float v8f;

#define D_MODEL   1024
#define N_EXP     64
#define N_TOKENS  32768
#define ROUTER_EPS 1e-6f

__global__ void router_zero_ws(float* __restrict__ ws) {
    if (threadIdx.x < N_EXP) ws[threadIdx.x] = 0.0f;
}

// 512 blocks x 128 threads (4 waves). Wave w handles tokens [blk*64 + w*16, +16).
__global__ __launch_bounds__(128) void router_main(const float* __restrict__ x,
                                                   const float* __restrict__ W,
                                                   float* __restrict__ out,
                                                   float* __restrict__ ws) {
    // padded logits tile: stride 68 -> lane l reads l*68+e, bank = (4l+e)%64, conflict-free
    __shared__ float s_logits[4][16][68];
    __shared__ float s_sums[N_EXP];

    const int tid  = threadIdx.x;
    const int wave = tid >> 5;
    const int lane = tid & 31;
    const int lo   = lane & 15;   // M (A rows) / N (B cols) index
    const int h    = lane >> 4;   // K half-select

    if (tid < N_EXP) s_sums[tid] = 0.0f;
    __syncthreads();

    const int tileTok = blockIdx.x * 64 + wave * 16;

    // --- GEMM: one wave -> 16 tokens x 64 experts, K=1024 ---------------------
    // A (16x4 f32): lane L holds A[M=L&15][k0+2h + j] in VGPR j  (ISA 7.12.2)
    // B ( 4x16 f32): lane L holds B[k0+2h + j][N=L&15] = W[N][k0+2h+j]
    const float* __restrict__ xrow = x + (size_t)(tileTok + lo) * D_MODEL;
    const float* __restrict__ wrow = W + (size_t)lo * D_MODEL;

    v8f acc0 = {}, acc1 = {}, acc2 = {}, acc3 = {};

#pragma unroll 4
    for (int k = 0; k < D_MODEL; k += 4) {
        const int kk = k + 2 * h;
        v2f a  = *(const v2f*)(xrow + kk);
        v2f b0 = *(const v2f*)(wrow + 0 * 16 * D_MODEL + kk);
        v2f b1 = *(const v2f*)(wrow + 1 * 16 * D_MODEL + kk);
        v2f b2 = *(const v2f*)(wrow + 2 * 16 * D_MODEL + kk);
        v2f b3 = *(const v2f*)(wrow + 3 * 16 * D_MODEL + kk);
        acc0 = __builtin_amdgcn_wmma_f32_16x16x4_f32(false, a, false, b0, (short)0, acc0, false, false);
        acc1 = __builtin_amdgcn_wmma_f32_16x16x4_f32(false, a, false, b1, (short)0, acc1, false, false);
        acc2 = __builtin_amdgcn_wmma_f32_16x16x4_f32(false, a, false, b2, (short)0, acc2, false, false);
        acc3 = __builtin_amdgcn_wmma_f32_16x16x4_f32(false, a, false, b3, (short)0, acc3, false, false);
    }

    // C/D layout: VGPR v holds (M = v + 8h, N = L&15). Spill to LDS by token row.
#pragma unroll
    for (int v = 0; v < 8; ++v) {
        const int tv = v + 8 * h;
        s_logits[wave][tv][ 0 + lo] = acc0[v];
        s_logits[wave][tv][16 + lo] = acc1[v];
        s_logits[wave][tv][32 + lo] = acc2[v];
        s_logits[wave][tv][48 + lo] = acc3[v];
    }
    __syncthreads();

    // --- per-token top-2 / gates / softmax-prob accumulation ------------------
    if (lane < 16) {
        const int tv = lane;
        const float* __restrict__ lg = &s_logits[wave][tv][0];

        float vals[N_EXP];
#pragma unroll
        for (int e = 0; e < N_EXP; ++e) vals[e] = lg[e];

        // ascending scan with strict '>' keeps lowest index on ties (jax top_k)
        float m1 = -FLT_MAX, m2 = -FLT_MAX;
        int   i1 = 0,        i2 = 0;
#pragma unroll
        for (int e = 0; e < N_EXP; ++e) {
            const float val = vals[e];
            if (val > m1)      { m2 = m1; i2 = i1; m1 = val; i1 = e; }
            else if (val > m2) { m2 = val; i2 = e; }
        }

        const float g1 = 1.0f / (1.0f + __expf(m2 - m1));  // softmax([m1,m2])[0]
        const float g2 = 1.0f - g1;

        const int t = tileTok + tv;
        out[2 * t + 0] = (float)i1;
        out[2 * t + 1] = (float)i2;
        out[2 * N_TOKENS + 2 * t + 0] = g1;
        out[2 * N_TOKENS + 2 * t + 1] = g2;

        float s = 0.0f;
#pragma unroll
        for (int e = 0; e < N_EXP; ++e) s += __expf(vals[e] - m1);
        const float inv = 1.0f / s;

        // staggered start -> 16 lanes hit 16 distinct counters each step
#pragma unroll
        for (int i = 0; i < N_EXP; ++i) {
            const int e = (i + tv * 4) & (N_EXP - 1);
            atomicAdd(&s_sums[e], __expf(vals[e] - m1) * inv);
        }
    }
    __syncthreads();

    if (tid < N_EXP) unsafeAtomicAdd(&ws[tid], s_sums[tid]);
}

// single block, 64 threads
__global__ void router_loss(const float* __restrict__ ws, float* __restrict__ out) {
    __shared__ float avg[N_EXP];
    const int e = threadIdx.x;
    if (e < N_EXP) avg[e] = ws[e] * (1.0f / (float)N_TOKENS);
    __syncthreads();
    if (e == 0) {
        float mean = 0.0f;
        for (int i = 0; i < N_EXP; ++i) mean += avg[i];
        mean *= (1.0f / N_EXP);
        float var = 0.0f;
        for (int i = 0; i < N_EXP; ++i) {
            const float d = avg[i] - mean;
            var += d * d;
        }
        var *= (1.0f / (N_EXP - 1));  // unbiased (ddof=1)
        const float denom = mean + ROUTER_EPS;
        out[4 * N_TOKENS] = var / (denom * denom);  // (std/denom)^2
    }
}

extern "C" void kernel_launch(void* const* d_in, const int* in_sizes, int n_in,
                              void* d_out, int out_size, void* d_ws, size_t ws_size,
                              hipStream_t stream) {
    (void)in_sizes; (void)n_in; (void)out_size; (void)ws_size;
    const float* x = (const float*)d_in[0];
    const float* W = (const float*)d_in[1];
    float* out = (float*)d_out;
    float* ws  = (float*)d_ws;

    router_zero_ws<<<1, 64, 0, stream>>>(ws);
    router_main<<<N_TOKENS / 64, 128, 0, stream>>>(x, W, out, ws);
    router_loss<<<1, 64, 0, stream>>>(ws, out);
}